// GCN_NoEdgeWeights_75118978007274
// MI455X (gfx1250) — compile-verified
//
#include <hip/hip_runtime.h>
#include <hip/hip_bf16.h>
#include <math.h>

typedef __attribute__((ext_vector_type(2))) float v2f;
typedef __attribute__((ext_vector_type(8))) float v8f;

#define HID 64

// ---------------- degree / dinv ----------------
__global__ void deg_count_kernel(const int* __restrict__ col, int* __restrict__ deg, int nE) {
  int e = blockIdx.x * blockDim.x + threadIdx.x;
  if (e < nE) atomicAdd(&deg[col[e]], 1);
}

__global__ void dinv_kernel(int* __restrict__ degbuf, int n) {
  int i = blockIdx.x * blockDim.x + threadIdx.x;
  if (i < n) {
    float d = (float)(degbuf[i] + 1);   // +1 self-loop -> deg >= 1 always
    ((float*)degbuf)[i] = rsqrtf(d);
  }
}

// ---------------- fp32 WMMA GEMM: Y = (X @ W) * scale[node] ----------------
// X: [n,64] row-major, W: [64,64] row-major, Y: [n,64].
// One wave computes one 16x16 output tile; 4 waves/block cover the 64 cols.
__global__ void gemm_wmma_scale(const float* __restrict__ X, const float* __restrict__ W,
                                const float* __restrict__ scale, float* __restrict__ Y, int n) {
  const int wave = threadIdx.x >> 5;       // 0..3 => column tile
  const int lane = threadIdx.x & 31;
  const int m    = lane & 15;              // row within tile (A), col within tile (B)
  const int kh   = (lane >> 4) << 1;       // K sub-pair: lanes 0-15 -> K0,K1 ; lanes 16-31 -> K2,K3
  const int row0 = blockIdx.x * 16;

  int arow = row0 + m; if (arow >= n) arow = n - 1;   // clamp, keep EXEC all-1s
  const float* xrow = X + (size_t)arow * HID;
  const int ncol = wave * 16 + m;

  v8f acc = {};
#pragma unroll
  for (int k = 0; k < HID; k += 4) {
    v2f a, b;
    a.x = xrow[k + kh];
    a.y = xrow[k + kh + 1];
    b.x = W[(k + kh) * HID + ncol];
    b.y = W[(k + kh + 1) * HID + ncol];
    acc = __builtin_amdgcn_wmma_f32_16x16x4_f32(
        /*neg_a=*/false, a, /*neg_b=*/false, b,
        /*c_mod=*/(short)0, acc, /*reuse_a=*/false, /*reuse_b=*/false);
  }

  // D layout: VGPR j -> M=j (lanes 0-15) / M=j+8 (lanes 16-31), N=lane%16
  const int mb = (lane >> 4) * 8;
#pragma unroll
  for (int j = 0; j < 8; ++j) {
    int r = row0 + mb + j;
    if (r < n) Y[(size_t)r * HID + ncol] = acc[j] * scale[r];
  }
}

// ---------------- edge scatter: agg[col] += y[row] ----------------
// 16 threads per edge, float4 each => coalesced 256B gather per edge.
__global__ void scatter_kernel(const int* __restrict__ row, const int* __restrict__ col,
                               const float* __restrict__ Y, float* __restrict__ agg, int nE) {
  unsigned t = blockIdx.x * blockDim.x + threadIdx.x;
  unsigned e = t >> 4;
  if (e >= (unsigned)nE) return;
  int sub = (t & 15) << 2;
  int r = row[e], c = col[e];
  const float4 v = *reinterpret_cast<const float4*>(Y + (size_t)r * HID + sub);
  float* dst = agg + (size_t)c * HID + sub;
  unsafeAtomicAdd(dst + 0, v.x);
  unsafeAtomicAdd(dst + 1, v.y);
  unsafeAtomicAdd(dst + 2, v.z);
  unsafeAtomicAdd(dst + 3, v.w);
}

// ---------------- finalize: h = relu(dinv*(agg + y) + bias) ----------------
__global__ void finalize_kernel(const float* __restrict__ Y, const float* __restrict__ agg,
                                const float* __restrict__ dinv, const float* __restrict__ bias,
                                float* __restrict__ Hout, int n) {
  unsigned t = blockIdx.x * blockDim.x + threadIdx.x;
  unsigned node = t >> 4;
  if (node >= (unsigned)n) return;
  int sub = (t & 15) << 2;
  float di = dinv[node];
  float4 a = *reinterpret_cast<const float4*>(agg + (size_t)node * HID + sub);
  float4 y = *reinterpret_cast<const float4*>(Y + (size_t)node * HID + sub);
  float4 b = *reinterpret_cast<const float4*>(bias + sub);
  float4 o;
  o.x = fmaxf(di * (a.x + y.x) + b.x, 0.0f);
  o.y = fmaxf(di * (a.y + y.y) + b.y, 0.0f);
  o.z = fmaxf(di * (a.z + y.z) + b.z, 0.0f);
  o.w = fmaxf(di * (a.w + y.w) + b.w, 0.0f);
  *reinterpret_cast<float4*>(Hout + (size_t)node * HID + sub) = o;
}

// ---------------- per-node head dots: p = h . wl[0:64], q = h . wl[64:128] ----------------
__global__ void pq_kernel(const float* __restrict__ Hm, const float* __restrict__ wl,
                          float* __restrict__ p, float* __restrict__ q, int n) {
  int node = blockIdx.x * (blockDim.x >> 5) + (threadIdx.x >> 5);
  int lane = threadIdx.x & 31;
  if (node >= n) return;
  const float* hr = Hm + (size_t)node * HID;
  float h0 = hr[lane], h1 = hr[32 + lane];
  float ps = h0 * wl[lane] + h1 * wl[32 + lane];
  float qs = h0 * wl[64 + lane] + h1 * wl[96 + lane];
#pragma unroll
  for (int off = 16; off > 0; off >>= 1) {
    ps += __shfl_xor(ps, off, 32);
    qs += __shfl_xor(qs, off, 32);
  }
  if (lane == 0) { p[node] = ps; q[node] = qs; }
}

// ---------------- edge head: sigmoid(p[row] + q[col] + b) ----------------
__global__ void edge_kernel(const int* __restrict__ row, const int* __restrict__ col,
                            const float* __restrict__ p, const float* __restrict__ q,
                            const float* __restrict__ blin, float* __restrict__ out, int nE) {
  int e = blockIdx.x * blockDim.x + threadIdx.x;
  if (e >= nE) return;
  float s = p[row[e]] + q[col[e]] + blin[0];
  out[e] = 1.0f / (1.0f + __expf(-s));
}

extern "C" void kernel_launch(void* const* d_in, const int* in_sizes, int n_in,
                              void* d_out, int out_size, void* d_ws, size_t ws_size,
                              hipStream_t stream) {
  const float* x  = (const float*)d_in[0];
  const int*   ei = (const int*)d_in[1];
  const float* w1 = (const float*)d_in[2];
  const float* b1 = (const float*)d_in[3];
  const float* w2 = (const float*)d_in[4];
  const float* b2 = (const float*)d_in[5];
  const float* wl = (const float*)d_in[6];
  const float* bl = (const float*)d_in[7];

  const int N = in_sizes[0] / HID;
  const int E = in_sizes[1] / 2;
  const int* row = ei;
  const int* col = ei + E;

  char* ws = (char*)d_ws;
  const size_t featBytes = (size_t)N * HID * sizeof(float);
  float* y    = (float*)(ws);
  float* agg  = (float*)(ws + featBytes);
  float* h    = (float*)(ws + 2 * featBytes);
  float* dinv = (float*)(ws + 3 * featBytes);
  float* p    = dinv + N;
  float* q    = p + N;

  const int B = 256;
  const unsigned gE   = (unsigned)((E + B - 1) / B);
  const unsigned gN   = (unsigned)((N + B - 1) / B);
  const unsigned gE16 = (unsigned)(((size_t)E * 16 + B - 1) / B);
  const unsigned gN16 = (unsigned)(((size_t)N * 16 + B - 1) / B);
  const unsigned gTile = (unsigned)((N + 15) / 16);

  // degree -> dinv (in place int->float)
  hipMemsetAsync(dinv, 0, (size_t)N * sizeof(int), stream);
  deg_count_kernel<<<gE, B, 0, stream>>>(col, (int*)dinv, E);
  dinv_kernel<<<gN, B, 0, stream>>>((int*)dinv, N);

  // ---- layer 1 ----
  gemm_wmma_scale<<<gTile, 128, 0, stream>>>(x, w1, dinv, y, N);
  hipMemsetAsync(agg, 0, featBytes, stream);
  scatter_kernel<<<gE16, B, 0, stream>>>(row, col, y, agg, E);
  finalize_kernel<<<gN16, B, 0, stream>>>(y, agg, dinv, b1, h, N);

  // ---- layer 2 ----
  gemm_wmma_scale<<<gTile, 128, 0, stream>>>(h, w2, dinv, y, N);
  hipMemsetAsync(agg, 0, featBytes, stream);
  scatter_kernel<<<gE16, B, 0, stream>>>(row, col, y, agg, E);
  finalize_kernel<<<gN16, B, 0, stream>>>(y, agg, dinv, b2, h, N);

  // ---- edge scoring head ----
  pq_kernel<<<(N + 7) / 8, B, 0, stream>>>(h, wl, p, q, N);
  edge_kernel<<<gE, B, 0, stream>>>(row, col, p, q, bl, (float*)d_out, E);
}